// GAT_55095840473360
// MI455X (gfx1250) — compile-verified
//
#include <hip/hip_runtime.h>
#include <hip/hip_bf16.h>

#define NN      100000
#define EE      1600000
#define ETOT    (EE + NN)      // edges + self loops
#define IN_DIM  128
#define HEADS   4
#define HID     15
#define HC      (HEADS * HID)  // 60
#define OUT_DIM 10
#define NEG_SLOPE 0.2f

typedef float v2f __attribute__((ext_vector_type(2)));
typedef float v8f __attribute__((ext_vector_type(8)));

// ---------- order-preserving float <-> uint encoding for atomic max ----------
__device__ __forceinline__ unsigned encf(float f) {
    unsigned u = __float_as_uint(f);
    return (u & 0x80000000u) ? ~u : (u | 0x80000000u);
}
__device__ __forceinline__ float decf(unsigned u) {
    u = (u & 0x80000000u) ? (u & 0x7FFFFFFFu) : ~u;
    return __uint_as_float(u);
}

// ---------- 1) h = x @ W via V_WMMA_F32_16X16X4_F32 ----------
// one wave per 16x16 output tile; 6250 row tiles x 4 col tiles (60 cols padded to 64).
// B columns >= 60 are loaded with a clamped (in-bounds) address and never masked:
// WMMA column n of C depends only on column n of B (lanes n, n+16), and we never
// store C columns >= 60, so the garbage stays confined to discarded lanes.
__global__ void k_gemm_wmma(const float* __restrict__ x, const float* __restrict__ W,
                            float* __restrict__ h) {
    const int tile    = blockIdx.x;
    const int rowTile = tile >> 2;
    const int colTile = tile & 3;
    const int lane    = threadIdx.x & 31;     // wave32
    const int m       = lane & 15;
    const int hi      = lane >> 4;            // 0 or 1
    const int kb      = hi * 2;               // K sub-offset per ISA A/B layout
    const int row     = rowTile * 16 + m;
    const int col     = colTile * 16 + m;
    const int colc    = col < HC ? col : HC - 1;   // address clamp only

    const float* __restrict__ xrow = x + (size_t)row * IN_DIM + kb;
    const float* __restrict__ wcol = W + (size_t)kb * HC + colc;

    v8f c = {};
    #pragma unroll
    for (int k = 0; k < IN_DIM; k += 4) {
        v2f a = *(const v2f*)(xrow + k);       // 8B-aligned -> global_load_b64
        v2f b;
        b.x = wcol[k * HC];                    // const offsets <= 30KB -> imm ioffset
        b.y = wcol[k * HC + HC];
        c = __builtin_amdgcn_wmma_f32_16x16x4_f32(
                /*neg_a=*/false, a, /*neg_b=*/false, b,
                /*c_mod=*/(short)0, c, /*reuse_a=*/false, /*reuse_b=*/false);
    }

    if (col < HC) {
        #pragma unroll
        for (int i = 0; i < 8; ++i) {
            int r = rowTile * 16 + i + hi * 8; // C layout: vgpr i -> M=i (+8 upper half)
            h[(size_t)r * HC + col] = c[i];
        }
    }
}

// ---------- 2) per-(node,head) attention coefficients ----------
__global__ void k_attn(const float* __restrict__ h,
                       const float* __restrict__ att_src, const float* __restrict__ att_dst,
                       float* __restrict__ a_src, float* __restrict__ a_dst) {
    int id = blockIdx.x * blockDim.x + threadIdx.x;
    if (id >= NN * HEADS) return;
    int n  = id >> 2;
    int hd = id & 3;
    const float* hp = h + (size_t)n * HC + hd * HID;
    const float* as = att_src + hd * HID;
    const float* ad = att_dst + hd * HID;
    float s0 = 0.f, s1 = 0.f;
    #pragma unroll
    for (int i = 0; i < HID; ++i) {
        float v = hp[i];
        s0 = fmaf(v, as[i], s0);
        s1 = fmaf(v, ad[i], s1);
    }
    a_src[id] = s0;
    a_dst[id] = s1;
}

__device__ __forceinline__ void edge_sd(const long long* __restrict__ ei, int eid,
                                        int& s, int& d) {
    if (eid < EE) { s = (int)ei[eid]; d = (int)ei[EE + eid]; }
    else          { s = d = eid - EE; }                        // self loop
}

__device__ __forceinline__ float edge_e(const float* __restrict__ a_src,
                                        const float* __restrict__ a_dst,
                                        int s, int d, int hd) {
    float e = a_src[s * HEADS + hd] + a_dst[d * HEADS + hd];
    return e < 0.f ? NEG_SLOPE * e : e;
}

// ---------- 3) segment max over incoming edges ----------
__global__ void k_edge_max(const long long* __restrict__ ei,
                           const float* __restrict__ a_src, const float* __restrict__ a_dst,
                           unsigned* __restrict__ mEnc) {
    int id = blockIdx.x * blockDim.x + threadIdx.x;
    if (id >= ETOT * HEADS) return;
    int eid = id >> 2, hd = id & 3;
    int s, d; edge_sd(ei, eid, s, d);
    float e = edge_e(a_src, a_dst, s, d, hd);
    atomicMax(&mEnc[d * HEADS + hd], encf(e));
}

// ---------- 4) segment sum of exp(e - max) ----------
__global__ void k_edge_denom(const long long* __restrict__ ei,
                             const float* __restrict__ a_src, const float* __restrict__ a_dst,
                             const unsigned* __restrict__ mEnc, float* __restrict__ denom) {
    int id = blockIdx.x * blockDim.x + threadIdx.x;
    if (id >= ETOT * HEADS) return;
    int eid = id >> 2, hd = id & 3;
    int s, d; edge_sd(ei, eid, s, d);
    float e = edge_e(a_src, a_dst, s, d, hd);
    atomicAdd(&denom[d * HEADS + hd], expf(e - decf(mEnc[d * HEADS + hd])));
}

// ---------- 5) message scatter: one 64-thread block per edge, lane = channel ----------
__global__ void k_scatter(const long long* __restrict__ ei,
                          const float* __restrict__ a_src, const float* __restrict__ a_dst,
                          const unsigned* __restrict__ mEnc, const float* __restrict__ denom,
                          const float* __restrict__ h, float* __restrict__ acc) {
    int eid = blockIdx.x;
    int c   = threadIdx.x;
    if (c >= HC) return;
    int s, d; edge_sd(ei, eid, s, d);
    int hd = c / HID;
    float e = edge_e(a_src, a_dst, s, d, hd);
    float alpha = expf(e - decf(mEnc[d * HEADS + hd])) / (denom[d * HEADS + hd] + 1e-16f);
    atomicAdd(&acc[(size_t)d * HC + c], alpha * h[(size_t)s * HC + c]);
}

// ---------- 6) bias + ELU + 60x10 linear + log_softmax ----------
__global__ void k_final(const float* __restrict__ acc, const float* __restrict__ bias,
                        const float* __restrict__ lin_w, const float* __restrict__ lin_b,
                        float* __restrict__ out) {
    int n = blockIdx.x * blockDim.x + threadIdx.x;
    if (n >= NN) return;
    float t[HC];
    const float4* a4 = (const float4*)(acc + (size_t)n * HC);   // 240B per node, 16B aligned
    #pragma unroll
    for (int i = 0; i < HC / 4; ++i) {
        float4 v = a4[i];
        t[4*i]   = v.x; t[4*i+1] = v.y; t[4*i+2] = v.z; t[4*i+3] = v.w;
    }
    #pragma unroll
    for (int ci = 0; ci < HC; ++ci) {
        float v = t[ci] + bias[ci];
        t[ci] = v > 0.f ? v : expm1f(v);                        // ELU(alpha=1)
    }
    float y[OUT_DIM];
    #pragma unroll
    for (int j = 0; j < OUT_DIM; ++j) y[j] = lin_b[j];
    for (int ci = 0; ci < HC; ++ci) {
        float tv = t[ci];
        #pragma unroll
        for (int j = 0; j < OUT_DIM; ++j)
            y[j] = fmaf(tv, lin_w[ci * OUT_DIM + j], y[j]);
    }
    float m = y[0];
    #pragma unroll
    for (int j = 1; j < OUT_DIM; ++j) m = fmaxf(m, y[j]);
    float ssum = 0.f;
    #pragma unroll
    for (int j = 0; j < OUT_DIM; ++j) ssum += expf(y[j] - m);
    float lse = m + logf(ssum);
    #pragma unroll
    for (int j = 0; j < OUT_DIM; ++j) out[(size_t)n * OUT_DIM + j] = y[j] - lse;
}

extern "C" void kernel_launch(void* const* d_in, const int* in_sizes, int n_in,
                              void* d_out, int out_size, void* d_ws, size_t ws_size,
                              hipStream_t stream) {
    const float*     x       = (const float*)d_in[0];
    const long long* ei      = (const long long*)d_in[1];   // int64 [2,E]
    const float*     W       = (const float*)d_in[2];
    const float*     att_src = (const float*)d_in[3];
    const float*     att_dst = (const float*)d_in[4];
    const float*     bias    = (const float*)d_in[5];
    const float*     lin_w   = (const float*)d_in[6];
    const float*     lin_b   = (const float*)d_in[7];
    float*           out     = (float*)d_out;

    // workspace layout (floats): h[N*60] | a_src[N*4] | a_dst[N*4] | mEnc[N*4] | denom[N*4] | acc[N*60]
    float*    h_buf = (float*)d_ws;
    float*    a_src = h_buf + (size_t)NN * HC;
    float*    a_dst = a_src + (size_t)NN * HEADS;
    unsigned* mEnc  = (unsigned*)(a_dst + (size_t)NN * HEADS);
    float*    denom = (float*)(mEnc + (size_t)NN * HEADS);
    float*    acc   = denom + (size_t)NN * HEADS;

    // zero the accumulator region (mEnc | denom | acc) — enc(f) > 0 for all reals, so 0 == -inf
    hipMemsetAsync(mEnc, 0, (size_t)NN * (HEADS + HEADS + HC) * sizeof(float), stream);

    // 1) GEMM (WMMA): 6250 row tiles x 4 col tiles, one wave each
    k_gemm_wmma<<<(NN / 16) * 4, 32, 0, stream>>>(x, W, h_buf);

    // 2) attention coefficients
    k_attn<<<(NN * HEADS + 255) / 256, 256, 0, stream>>>(h_buf, att_src, att_dst, a_src, a_dst);

    // 3) segment max
    k_edge_max<<<(ETOT * HEADS + 255) / 256, 256, 0, stream>>>(ei, a_src, a_dst, mEnc);

    // 4) segment softmax denominator
    k_edge_denom<<<(ETOT * HEADS + 255) / 256, 256, 0, stream>>>(ei, a_src, a_dst, mEnc, denom);

    // 5) weighted message scatter (one block of 64 lanes per edge)
    k_scatter<<<ETOT, 64, 0, stream>>>(ei, a_src, a_dst, mEnc, denom, h_buf, acc);

    // 6) finalize per node
    k_final<<<(NN + 127) / 128, 128, 0, stream>>>(acc, bias, lin_w, lin_b, out);
}